// AAALinear_40192303956806
// MI455X (gfx1250) — compile-verified
//
#include <hip/hip_runtime.h>
#include <math.h>

typedef __attribute__((ext_vector_type(2))) float v2f;
typedef __attribute__((ext_vector_type(8))) float v8f;

#define N_ROWS 16000
#define K_DIM  512
#define C_DIM  1000
#define NT_N   63            // ceil(1000/16)
#define NITER  100
#define LR     0.1f
#define CALW   5.0f
#define B1     0.9f
#define B2     0.999f
#define EPS    1e-8f
#define INTERVAL 4.0f
#define DEV    0.5f
#define EPT    32            // 32 lanes * 32 elements = 1024 >= 1000

// ---------------------------------------------------------------------------
// Kernel 1: logits = x @ W via V_WMMA_F32_16X16X4_F32. One wave per 16x16 tile.
// Out-of-range N columns handled by CLAMPING the load address: garbage B values
// only pollute output columns that are never stored -> no inner-loop predication.
// ---------------------------------------------------------------------------
__global__ __launch_bounds__(256) void gemm_f32_wmma(const float* __restrict__ x,
                                                     const float* __restrict__ W,
                                                     float* __restrict__ out) {
  const int lane = threadIdx.x & 31;
  const int wave = threadIdx.x >> 5;
  const int tile = blockIdx.x * 8 + wave;      // 63000 tiles total, wave-uniform
  const int tm = tile / NT_N;                  // 0..999
  const int tn = tile % NT_N;                  // 0..62

  const int half = lane >> 4;                  // 0: K pair {0,1}, 1: K pair {2,3}
  const int l15  = lane & 15;
  const int rowA = tm * 16 + l15;              // A: lanes hold one M row each
  const int colB = tn * 16 + l15;              // B: lanes hold one N col each
  const int colS = (colB < C_DIM) ? colB : (C_DIM - 1);  // clamped, always legal

  const float* __restrict__ pa = x + (size_t)rowA * K_DIM + half * 2;
  const float* __restrict__ pb = W + (size_t)(half * 2) * C_DIM + colS;

  v8f acc = {};
#pragma unroll 4
  for (int k = 0; k < K_DIM; k += 4) {
    v2f a;                                     // A 16x4: VGPR0=K ka, VGPR1=K ka+1
    a.x = pa[0];
    a.y = pa[1];
    v2f b;                                     // B 4x16: row = vgpr + 2*half
    b.x = pb[0];
    b.y = pb[C_DIM];
    acc = __builtin_amdgcn_wmma_f32_16x16x4_f32(false, a, false, b,
                                                (short)0, acc, false, false);
    pa += 4;
    pb += 4 * C_DIM;
  }
  // C/D 16x16: VGPR r -> row (half*8 + r), col = l15
  const int crow = tm * 16 + half * 8;
  if (colB < C_DIM) {
#pragma unroll
    for (int r = 0; r < 8; ++r)
      out[(size_t)(crow + r) * C_DIM + colB] = acc[r];
  }
}

// ---------------------------------------------------------------------------
// Kernel 2: one wave32 per row; all state in registers; reductions are pure
// shfl_xor butterflies (no LDS, no barriers). p_first tracked analytically.
// ---------------------------------------------------------------------------
__device__ __forceinline__ float signf(float x) {
  return (x > 0.0f) ? 1.0f : ((x < 0.0f) ? -1.0f : 0.0f);
}

// butterfly (value,index) argmax, min-index tie-break; result in all lanes
__device__ __forceinline__ void wave_argmax(float v, int i, float& oV, int& oI) {
#pragma unroll
  for (int off = 16; off >= 1; off >>= 1) {
    float ov = __shfl_xor(v, off, 32);
    int   oi = __shfl_xor(i, off, 32);
    if (ov > v || (ov == v && oi < i)) { v = ov; i = oi; }
  }
  oV = v; oI = i;
}

__device__ __forceinline__ float wave_sum(float s) {
#pragma unroll
  for (int off = 16; off >= 1; off >>= 1) s += __shfl_xor(s, off, 32);
  return s;
}

__global__ __launch_bounds__(256) void adam_attractor(float* __restrict__ logits) {
  const int lane = threadIdx.x & 31;
  const int wid  = threadIdx.x >> 5;
  const int row  = blockIdx.x * 8 + wid;       // grid covers exactly 16000 rows
  float* Lrow    = logits + (size_t)row * C_DIM;

  float L[EPT], m[EPT], v[EPT], e[EPT];
#pragma unroll
  for (int i = 0; i < EPT; ++i) {
    const int j = lane + i * 32;               // coalesced per instruction
    L[i] = (j < C_DIM) ? Lrow[j] : -INFINITY;  // pads never win / contribute 0
    m[i] = 0.0f; v[i] = 0.0f;
  }

  // ---- fixed per-row quantities from original logits ----
  float lv = -INFINITY; int li = 0x7fffffff;
#pragma unroll
  for (int i = 0; i < EPT; ++i) {
    const int j = lane + i * 32;
    if (L[i] > lv || (L[i] == lv && j < li)) { lv = L[i]; li = j; }
  }
  float v0o; int idx0;
  wave_argmax(lv, li, v0o, idx0);

  lv = -INFINITY; li = 0x7fffffff;
#pragma unroll
  for (int i = 0; i < EPT; ++i) {
    const int j = lane + i * 32;
    const float val = (j == idx0) ? -INFINITY : L[i];
    if (val > lv || (val == lv && j < li)) { lv = val; li = j; }
  }
  float v1o; int dummy;
  wave_argmax(lv, li, v1o, dummy);

  float ls = 0.0f;
#pragma unroll
  for (int i = 0; i < EPT; ++i) ls += __expf(L[i] - v0o);
  const float sum0 = wave_sum(ls);

  const float pmax_ori   = 1.0f / sum0;                 // exp(max-max)/sum
  const float margin_ori = v0o - v1o;
  const float attractor  = (rintf(margin_ori / INTERVAL + DEV) - DEV) * INTERVAL;
  const float target     = 2.0f * attractor - margin_ori;   // REVERSE = 1
  const float invBS      = 1.0f / 1000.0f;

  // wave-uniform shadow of element idx0 (idx0 is the original argmax, so its
  // current value starts at v0o). Updated with arithmetic identical to the
  // per-element path, so it stays bit-exact with L[idx0].
  float Li0 = v0o, mi0 = 0.0f, vi0 = 0.0f;

  // ---- Adam loop ----
  float b1p = B1, b2p = B2;                             // B1^t, B2^t
  for (int t = 0; t < NITER; ++t) {
    // top-1
    lv = -INFINITY; li = 0x7fffffff;
#pragma unroll
    for (int i = 0; i < EPT; ++i) {
      const int j = lane + i * 32;
      if (L[i] > lv || (L[i] == lv && j < li)) { lv = L[i]; li = j; }
    }
    float vmax; int a0;
    wave_argmax(lv, li, vmax, a0);

    // top-2 (exclude a0)
    lv = -INFINITY; li = 0x7fffffff;
#pragma unroll
    for (int i = 0; i < EPT; ++i) {
      const int j = lane + i * 32;
      const float val = (j == a0) ? -INFINITY : L[i];
      if (val > lv || (val == lv && j < li)) { lv = val; li = j; }
    }
    float v1c; int a1;
    wave_argmax(lv, li, v1c, a1);

    // softmax pieces
    ls = 0.0f;
#pragma unroll
    for (int i = 0; i < EPT; ++i) { e[i] = __expf(L[i] - vmax); ls += e[i]; }
    const float sumexp  = wave_sum(ls);
    const float inv_sum = 1.0f / sumexp;
    const float p_first = __expf(Li0 - vmax) * inv_sum;   // == e[idx0]*inv_sum

    const float sdef = signf((vmax - v1c) - target) * invBS;
    const float ccal = CALW * signf(p_first - pmax_ori) * p_first * invBS;

    const float bc1 = 1.0f / (1.0f - b1p);
    const float bc2 = 1.0f / (1.0f - b2p);
#pragma unroll
    for (int i = 0; i < EPT; ++i) {
      const int j = lane + i * 32;
      float g = ccal * (((j == idx0) ? 1.0f : 0.0f) - e[i] * inv_sum);
      g += sdef * (((j == a0) ? 1.0f : 0.0f) - ((j == a1) ? 1.0f : 0.0f));
      m[i] = B1 * m[i] + (1.0f - B1) * g;
      v[i] = B2 * v[i] + (1.0f - B2) * g * g;
      L[i] -= LR * (m[i] * bc1) / (sqrtf(v[i] * bc2) + EPS);
    }
    // identical update for the wave-uniform idx0 shadow
    {
      float g = ccal * (1.0f - __expf(Li0 - vmax) * inv_sum);
      g += sdef * (((idx0 == a0) ? 1.0f : 0.0f) - ((idx0 == a1) ? 1.0f : 0.0f));
      mi0 = B1 * mi0 + (1.0f - B1) * g;
      vi0 = B2 * vi0 + (1.0f - B2) * g * g;
      Li0 -= LR * (mi0 * bc1) / (sqrtf(vi0 * bc2) + EPS);
    }
    b1p *= B1; b2p *= B2;
  }

#pragma unroll
  for (int i = 0; i < EPT; ++i) {
    const int j = lane + i * 32;
    if (j < C_DIM) Lrow[j] = L[i];
  }
}

// ---------------------------------------------------------------------------
extern "C" void kernel_launch(void* const* d_in, const int* in_sizes, int n_in,
                              void* d_out, int out_size, void* d_ws, size_t ws_size,
                              hipStream_t stream) {
  const float* x = (const float*)d_in[0];   // 16000 x 512
  const float* W = (const float*)d_in[1];   // 512 x 1000
  float* out = (float*)d_out;               // 16000 x 1000

  const int tiles  = (N_ROWS / 16) * NT_N;  // 63000
  const int blocks = tiles / 8;             // 7875 (exact)
  gemm_f32_wmma<<<blocks, 256, 0, stream>>>(x, W, out);
  adam_attractor<<<N_ROWS / 8, 256, 0, stream>>>(out);   // 1 wave per row
}